// GNNBaseLayer_71648644432061
// MI455X (gfx1250) — compile-verified
//
#include <hip/hip_runtime.h>
#include <hip/hip_bf16.h>
#include <math.h>

// ---------------------------------------------------------------------------
// GNN base layer fused into WMMA kernels for gfx1250 (CDNA5, wave32).
//   K0: fold BN into weights (msg + upd), compute effective biases.
//   K1: zero segment sums / counts.
//   K2: per-edge  msg = gelu(BN(node[dst])@Wm+bm) * gelu(edge@We+be),
//       atomic scatter-add into sums[src], counts[src].   (WMMA f32 16x16x4)
//   K3: per-node  out = gelu(BN(concat(node, sums/cnt))@Wu+bu). (WMMA)
// ---------------------------------------------------------------------------

#define N_NODES 50000
#define N_EDGES 800000
#define DD 64
#define HH 64
#define CC 128
#define BN_EPS 1e-3f
#define LDW 65   // padded LDS row stride (floats) to avoid bank conflicts

typedef __attribute__((ext_vector_type(2))) float v2f;
typedef __attribute__((ext_vector_type(8))) float v8f;

static __device__ __forceinline__ v8f wmma_f32(v2f a, v2f b, v8f c) {
  // D(16x16,f32) = A(16x4,f32) x B(4x16,f32) + C
  return __builtin_amdgcn_wmma_f32_16x16x4_f32(false, a, false, b,
                                               (short)0, c, false, false);
}

static __device__ __forceinline__ float gelu_exact(float x) {
  return 0.5f * x * (1.0f + erff(x * 0.70710678118654752440f));
}

// ---------------------------------------------------------------------------
// K0: fold BN scale/shift into W and b.
//   scale[k] = gamma[k]*rsqrt(var[k]+eps); shift[k] = beta[k]-mean[k]*scale[k]
//   Wf[k][j] = scale[k]*W[k][j];  bf[j] = b[j] + sum_k shift[k]*W[k][j]
// ---------------------------------------------------------------------------
__global__ void fold_bn_kernel(
    const float* __restrict__ msg_gamma, const float* __restrict__ msg_beta,
    const float* __restrict__ msg_mean,  const float* __restrict__ msg_var,
    const float* __restrict__ msg_W,     const float* __restrict__ msg_b,
    const float* __restrict__ upd_gamma, const float* __restrict__ upd_beta,
    const float* __restrict__ upd_mean,  const float* __restrict__ upd_var,
    const float* __restrict__ upd_W,     const float* __restrict__ upd_b,
    float* __restrict__ msg_Wf, float* __restrict__ msg_bf,
    float* __restrict__ upd_Wf, float* __restrict__ upd_bf) {
  __shared__ float sc[CC], sh[CC];
  const int t = threadIdx.x;
  if (t < DD) {
    float s = msg_gamma[t] * rsqrtf(msg_var[t] + BN_EPS);
    sc[t] = s;
    sh[t] = msg_beta[t] - msg_mean[t] * s;
  }
  __syncthreads();
  for (int i = t; i < DD * HH; i += blockDim.x)
    msg_Wf[i] = sc[i >> 6] * msg_W[i];
  if (t < HH) {
    float acc = msg_b[t];
    for (int k = 0; k < DD; ++k) acc += sh[k] * msg_W[k * HH + t];
    msg_bf[t] = acc;
  }
  __syncthreads();
  if (t < CC) {
    float s = upd_gamma[t] * rsqrtf(upd_var[t] + BN_EPS);
    sc[t] = s;
    sh[t] = upd_beta[t] - upd_mean[t] * s;
  }
  __syncthreads();
  for (int i = t; i < CC * HH; i += blockDim.x)
    upd_Wf[i] = sc[i >> 6] * upd_W[i];
  if (t < HH) {
    float acc = upd_b[t];
    for (int k = 0; k < CC; ++k) acc += sh[k] * upd_W[k * HH + t];
    upd_bf[t] = acc;
  }
}

// ---------------------------------------------------------------------------
// K1: zero the segment-sum + count workspace.
// ---------------------------------------------------------------------------
__global__ void zero_kernel(float* __restrict__ p, int n) {
  int i = blockIdx.x * blockDim.x + threadIdx.x;
  if (i < n) p[i] = 0.0f;
}

// ---------------------------------------------------------------------------
// K2: edge kernel. One wave per 16-edge tile (grid-strided).
//   acc1 = node[dst]@msg_Wf + msg_bf  (BN pre-folded)
//   acc2 = edge       @edge_W + edge_b
//   scatter gelu(acc1)*gelu(acc2) into sums[src], +1 into cnt[src].
// ---------------------------------------------------------------------------
__global__ void __launch_bounds__(256) edge_kernel(
    const float* __restrict__ node_feats, const float* __restrict__ edge_feats,
    const int* __restrict__ src_idx, const int* __restrict__ dst_idx,
    const float* __restrict__ msg_Wf, const float* __restrict__ msg_bf,
    const float* __restrict__ edge_W, const float* __restrict__ edge_b,
    float* __restrict__ sums, float* __restrict__ cnt) {
  __shared__ float lWm[DD * LDW];
  __shared__ float lWe[DD * LDW];
  __shared__ float lbm[HH];
  __shared__ float lbe[HH];

  for (int i = threadIdx.x; i < DD * HH; i += blockDim.x) {
    int k = i >> 6, j = i & 63;
    lWm[k * LDW + j] = msg_Wf[i];
    lWe[k * LDW + j] = edge_W[i];
  }
  if (threadIdx.x < HH) {
    lbm[threadIdx.x] = msg_bf[threadIdx.x];
    lbe[threadIdx.x] = edge_b[threadIdx.x];
  }
  __syncthreads();

  const int lane = threadIdx.x & 31;
  const int half = lane >> 4;   // 0 or 1: K/M split of the wave
  const int lrow = lane & 15;
  const int waveId = (blockIdx.x * blockDim.x + threadIdx.x) >> 5;
  const int nWaves = (gridDim.x * blockDim.x) >> 5;
  const int nTiles = N_EDGES / 16;

  for (int tile = waveId; tile < nTiles; tile += nWaves) {
    const int e0 = tile * 16;
    // per-lane A rows: row = lrow (both lane halves), K offset 2*half
    const int drow = dst_idx[e0 + lrow];
    const float* nrow = node_feats + (size_t)drow * DD + 2 * half;
    const float* erow = edge_feats + (size_t)(e0 + lrow) * DD + 2 * half;
    v2f a1[16], a2[16];
#pragma unroll
    for (int k = 0; k < 16; ++k) {
      a1[k] = *(const v2f*)(nrow + 4 * k);
      a2[k] = *(const v2f*)(erow + 4 * k);
    }
    int srcs[8];
#pragma unroll
    for (int v = 0; v < 8; ++v) srcs[v] = src_idx[e0 + v + 8 * half];
    if (lane < 16) atomicAdd(&cnt[src_idx[e0 + lane]], 1.0f);

#pragma unroll
    for (int nt = 0; nt < 4; ++nt) {
      const int col = nt * 16 + lrow;
      const float b1 = lbm[col], b2 = lbe[col];
      v8f acc1 = {b1, b1, b1, b1, b1, b1, b1, b1};
      v8f acc2 = {b2, b2, b2, b2, b2, b2, b2, b2};
#pragma unroll
      for (int k = 0; k < 16; ++k) {
        const int kk = 4 * k + 2 * half;
        v2f bw1 = {lWm[kk * LDW + col], lWm[(kk + 1) * LDW + col]};
        acc1 = wmma_f32(a1[k], bw1, acc1);
        v2f bw2 = {lWe[kk * LDW + col], lWe[(kk + 1) * LDW + col]};
        acc2 = wmma_f32(a2[k], bw2, acc2);
      }
#pragma unroll
      for (int v = 0; v < 8; ++v) {
        const float m = gelu_exact(acc1[v]) * gelu_exact(acc2[v]);
        atomicAdd(&sums[(size_t)srcs[v] * HH + col], m);
      }
    }
  }
}

// ---------------------------------------------------------------------------
// K3: node kernel. One wave per 16-node tile (grid-strided).
//   A = [node_feats[n] , sums[n]/max(cnt[n],1)]  (K = 128, BN pre-folded)
//   out = gelu(A @ upd_Wf + upd_bf)
// ---------------------------------------------------------------------------
__global__ void __launch_bounds__(256) node_kernel(
    const float* __restrict__ node_feats, const float* __restrict__ sums,
    const float* __restrict__ cnt, const float* __restrict__ upd_Wf,
    const float* __restrict__ upd_bf, float* __restrict__ out) {
  __shared__ float lWu[CC * LDW];
  __shared__ float lbu[HH];

  for (int i = threadIdx.x; i < CC * HH; i += blockDim.x) {
    int k = i >> 6, j = i & 63;
    lWu[k * LDW + j] = upd_Wf[i];
  }
  if (threadIdx.x < HH) lbu[threadIdx.x] = upd_bf[threadIdx.x];
  __syncthreads();

  const int lane = threadIdx.x & 31;
  const int half = lane >> 4;
  const int lrow = lane & 15;
  const int waveId = (blockIdx.x * blockDim.x + threadIdx.x) >> 5;
  const int nWaves = (gridDim.x * blockDim.x) >> 5;
  const int nTiles = N_NODES / 16;  // 50000/16 = 3125 exactly

  for (int tile = waveId; tile < nTiles; tile += nWaves) {
    const int row = tile * 16 + lrow;
    const float inv = 1.0f / fmaxf(cnt[row], 1.0f);
    const float* nrow = node_feats + (size_t)row * DD + 2 * half;
    const float* srow = sums + (size_t)row * HH + 2 * half;
    v2f a[32];
#pragma unroll
    for (int k = 0; k < 16; ++k) a[k] = *(const v2f*)(nrow + 4 * k);
#pragma unroll
    for (int k = 0; k < 16; ++k) {
      v2f t = *(const v2f*)(srow + 4 * k);
      t.x *= inv;
      t.y *= inv;
      a[16 + k] = t;
    }
#pragma unroll
    for (int nt = 0; nt < 4; ++nt) {
      const int col = nt * 16 + lrow;
      const float b = lbu[col];
      v8f acc = {b, b, b, b, b, b, b, b};
#pragma unroll
      for (int k = 0; k < 32; ++k) {
        const int kk = 4 * k + 2 * half;
        v2f bw = {lWu[kk * LDW + col], lWu[(kk + 1) * LDW + col]};
        acc = wmma_f32(a[k], bw, acc);
      }
#pragma unroll
      for (int v = 0; v < 8; ++v) {
        const int m = tile * 16 + v + 8 * half;
        out[(size_t)m * HH + col] = gelu_exact(acc[v]);
      }
    }
  }
}

// ---------------------------------------------------------------------------
extern "C" void kernel_launch(void* const* d_in, const int* in_sizes, int n_in,
                              void* d_out, int out_size, void* d_ws,
                              size_t ws_size, hipStream_t stream) {
  const float* node_feats = (const float*)d_in[0];
  const float* edge_feats = (const float*)d_in[1];
  const int* src_idx = (const int*)d_in[2];
  const int* dst_idx = (const int*)d_in[3];
  const float* msg_gamma = (const float*)d_in[4];
  const float* msg_beta = (const float*)d_in[5];
  const float* msg_mean = (const float*)d_in[6];
  const float* msg_var = (const float*)d_in[7];
  const float* msg_W = (const float*)d_in[8];
  const float* msg_b = (const float*)d_in[9];
  const float* edge_W = (const float*)d_in[10];
  const float* edge_b = (const float*)d_in[11];
  const float* upd_gamma = (const float*)d_in[12];
  const float* upd_beta = (const float*)d_in[13];
  const float* upd_mean = (const float*)d_in[14];
  const float* upd_var = (const float*)d_in[15];
  const float* upd_W = (const float*)d_in[16];
  const float* upd_b = (const float*)d_in[17];

  // workspace layout (floats)
  float* sums = (float*)d_ws;                      // N*H
  float* cnt = sums + (size_t)N_NODES * HH;        // N
  float* msg_Wf = cnt + N_NODES;                   // 64*64
  float* msg_bf = msg_Wf + DD * HH;                // 64
  float* upd_Wf = msg_bf + HH;                     // 128*64
  float* upd_bf = upd_Wf + CC * HH;                // 64

  fold_bn_kernel<<<1, 256, 0, stream>>>(
      msg_gamma, msg_beta, msg_mean, msg_var, msg_W, msg_b,
      upd_gamma, upd_beta, upd_mean, upd_var, upd_W, upd_b,
      msg_Wf, msg_bf, upd_Wf, upd_bf);

  const int nzero = N_NODES * HH + N_NODES;
  zero_kernel<<<(nzero + 255) / 256, 256, 0, stream>>>(sums, nzero);

  edge_kernel<<<800, 256, 0, stream>>>(node_feats, edge_feats, src_idx,
                                       dst_idx, msg_Wf, msg_bf, edge_W, edge_b,
                                       sums, cnt);

  node_kernel<<<400, 256, 0, stream>>>(node_feats, sums, cnt, upd_Wf, upd_bf,
                                       (float*)d_out);
}